// MultiPerspectiveMatch_43920335569191
// MI455X (gfx1250) — compile-verified
//
#include <hip/hip_runtime.h>
#include <math.h>

#define L   200   // Lq == Ld
#define KD  200   // feature dim K
#define KP  201   // padded LDS row stride (conflict-free: gcd(201,64)=1 -> 9n mod 64 distinct)
#define C   50    // channels
#define NB  32    // batch
#define NT  13    // ceil(200/16) tiles
#define EPSF 1e-6f
#define OUTC 250  // 5*C

typedef float v2f __attribute__((ext_vector_type(2)));
typedef float v8f __attribute__((ext_vector_type(8)));

static __device__ __forceinline__ v8f wmma_f32(v2f a, v2f b, v8f c) {
  // V_WMMA_F32_16X16X4_F32 : D = A(16x4 f32) x B(4x16 f32) + C(16x16 f32)
  return __builtin_amdgcn_wmma_f32_16x16x4_f32(false, a, false, b, (short)0, c,
                                               false, false);
}

static __device__ __forceinline__ int imin(int a, int b) { return a < b ? a : b; }
static __device__ __forceinline__ float rnorm(float s) {
  return 1.0f / sqrtf(fmaxf(s, EPSF));  // reciprocal norm
}

// ---------------------------------------------------------------------------
// Kernel 1a: d-row reciprocal norms: rdn[b,i] + 4 per-channel sets.
// ---------------------------------------------------------------------------
__global__ __launch_bounds__(64) void d_norms_kernel(
    const float* __restrict__ d_rep,
    const float* __restrict__ fM, const float* __restrict__ mM,
    const float* __restrict__ aM, const float* __restrict__ xM,
    float* __restrict__ rdn, float* __restrict__ rfpn, float* __restrict__ rmpn,
    float* __restrict__ rapn, float* __restrict__ rxpn) {
  const int bi = blockIdx.x;  // b*L + i
  const float* row = d_rep + (size_t)bi * KD;
  __shared__ float xsh[KD];
  __shared__ float red[64];
  const int tid = threadIdx.x;
  float ss = 0.f;
  for (int k = tid; k < KD; k += 64) {
    float v = row[k];
    xsh[k] = v;
    ss += v * v;
  }
  red[tid] = ss;
  __syncthreads();
  if (tid == 0) {
    float s = 0.f;
    for (int t = 0; t < 64; ++t) s += red[t];
    rdn[bi] = rnorm(s);
  }
  if (tid < C) {
    const int c = tid;
    const float* fr = fM + (size_t)c * KD;
    const float* mr = mM + (size_t)c * KD;
    const float* ar = aM + (size_t)c * KD;
    const float* xr = xM + (size_t)c * KD;
    float af = 0.f, amm = 0.f, aa = 0.f, ax = 0.f;
    for (int k = 0; k < KD; ++k) {
      float x2 = xsh[k] * xsh[k];
      float m;
      m = fr[k]; af  += x2 * m * m;
      m = mr[k]; amm += x2 * m * m;
      m = ar[k]; aa  += x2 * m * m;
      m = xr[k]; ax  += x2 * m * m;
    }
    size_t o = (size_t)bi * C + c;
    rfpn[o] = rnorm(af);
    rmpn[o] = rnorm(amm);
    rapn[o] = rnorm(aa);
    rxpn[o] = rnorm(ax);
  }
}

// ---------------------------------------------------------------------------
// Kernel 1b: q-row reciprocal norms: rqn[b,j], maxpool + max_att sets.
// ---------------------------------------------------------------------------
__global__ __launch_bounds__(64) void q_norms_kernel(
    const float* __restrict__ q_rep,
    const float* __restrict__ mM, const float* __restrict__ xM,
    float* __restrict__ rqn, float* __restrict__ rmqn,
    float* __restrict__ rxqn) {
  const int bj = blockIdx.x;
  const float* row = q_rep + (size_t)bj * KD;
  __shared__ float xsh[KD];
  __shared__ float red[64];
  const int tid = threadIdx.x;
  float ss = 0.f;
  for (int k = tid; k < KD; k += 64) {
    float v = row[k];
    xsh[k] = v;
    ss += v * v;
  }
  red[tid] = ss;
  __syncthreads();
  if (tid == 0) {
    float s = 0.f;
    for (int t = 0; t < 64; ++t) s += red[t];
    rqn[bj] = rnorm(s);
  }
  if (tid < C) {
    const int c = tid;
    const float* mr = mM + (size_t)c * KD;
    const float* xr = xM + (size_t)c * KD;
    float amm = 0.f, ax = 0.f;
    for (int k = 0; k < KD; ++k) {
      float x2 = xsh[k] * xsh[k];
      float m;
      m = mr[k]; amm += x2 * m * m;
      m = xr[k]; ax  += x2 * m * m;
    }
    size_t o = (size_t)bj * C + c;
    rmqn[o] = rnorm(amm);
    rxqn[o] = rnorm(ax);
  }
}

// ---------------------------------------------------------------------------
// Kernel 1c: q_last reciprocal channel norms under full_M.
// ---------------------------------------------------------------------------
__global__ __launch_bounds__(64) void qlast_norms_kernel(
    const float* __restrict__ q_last, const float* __restrict__ fM,
    float* __restrict__ rfqn) {
  const int b = blockIdx.x;
  const float* row = q_last + (size_t)b * KD;
  __shared__ float xsh[KD];
  const int tid = threadIdx.x;
  for (int k = tid; k < KD; k += 64) xsh[k] = row[k];
  __syncthreads();
  if (tid < C) {
    const int c = tid;
    const float* fr = fM + (size_t)c * KD;
    float acc = 0.f;
    for (int k = 0; k < KD; ++k) {
      float m = fr[k];
      acc += xsh[k] * xsh[k] * m * m;
    }
    rfqn[(size_t)b * C + c] = rnorm(acc);
  }
}

// ---------------------------------------------------------------------------
// Kernel 2: rel[b,i,j] = (d_i . q_j) * rqn_j * rdn_i * masks via WMMA.
// One wave per 16x16 (i,j) tile.
// ---------------------------------------------------------------------------
__global__ __launch_bounds__(32) void rel_kernel(
    const float* __restrict__ d_rep, const float* __restrict__ q_rep,
    const float* __restrict__ q_mask, const float* __restrict__ d_mask,
    const float* __restrict__ rqn, const float* __restrict__ rdn,
    float* __restrict__ rel) {
  const int b = blockIdx.y;
  const int ti = blockIdx.x / NT, tj = blockIdx.x % NT;
  const int i0 = ti * 16, j0 = tj * 16;
  const int lane = threadIdx.x;
  const int half = lane >> 4, m = lane & 15;
  const int ia = imin(i0 + m, L - 1);
  const int ja = imin(j0 + m, L - 1);
  const float* drow = d_rep + ((size_t)b * L + ia) * KD;
  const float* qrow = q_rep + ((size_t)b * L + ja) * KD;
  v8f acc = {};
  for (int k0 = 0; k0 < KD; k0 += 4) {
    const int kk = k0 + half * 2;
    v2f a, bb;
    a.x = drow[kk];  a.y = drow[kk + 1];
    bb.x = qrow[kk]; bb.y = qrow[kk + 1];
    acc = wmma_f32(a, bb, acc);
  }
  for (int v = 0; v < 8; ++v) {
    const int i = i0 + v + half * 8;
    const int j = j0 + m;
    if (i < L && j < L) {
      float val = acc[v] * rqn[b * L + j] * rdn[b * L + i];
      val *= q_mask[b * L + j] * d_mask[b * L + i];
      rel[((size_t)b * L + i) * L + j] = val;
    }
  }
}

// ---------------------------------------------------------------------------
// Kernel 3: per (b,i): 1/(sum_j rel + EPS) and argmax_j rel.
// ---------------------------------------------------------------------------
__global__ __launch_bounds__(64) void rel_reduce_kernel(
    const float* __restrict__ rel, float* __restrict__ rsuminv,
    int* __restrict__ argidx) {
  const int bi = blockIdx.x;
  const float* row = rel + (size_t)bi * L;
  __shared__ float s_sum[64], s_max[64];
  __shared__ int s_idx[64];
  const int tid = threadIdx.x;
  float sm = 0.f, mx = -3.402823466e38f;
  int mi = 0;
  for (int j = tid; j < L; j += 64) {
    float v = row[j];
    sm += v;
    if (v > mx) { mx = v; mi = j; }
  }
  s_sum[tid] = sm; s_max[tid] = mx; s_idx[tid] = mi;
  __syncthreads();
  if (tid == 0) {
    float tot = 0.f, bm = -3.402823466e38f;
    int bidx = 0;
    for (int t = 0; t < 64; ++t) {
      tot += s_sum[t];
      if (s_max[t] > bm) { bm = s_max[t]; bidx = s_idx[t]; }
    }
    rsuminv[bi] = 1.0f / (tot + EPSF);
    argidx[bi] = bidx;
  }
}

// ---------------------------------------------------------------------------
// Kernel 4: wq[b,i,k] = (rel[b,i,:] . q[b,:,k]) * rsuminv[b,i] via WMMA.
// One wave per 16x16 (i, k) tile; contraction over j.
// ---------------------------------------------------------------------------
__global__ __launch_bounds__(32) void wq_kernel(
    const float* __restrict__ rel, const float* __restrict__ q_rep,
    const float* __restrict__ rsuminv, float* __restrict__ wq) {
  const int b = blockIdx.y;
  const int ti = blockIdx.x / NT, tk = blockIdx.x % NT;
  const int i0 = ti * 16, k0t = tk * 16;
  const int lane = threadIdx.x;
  const int half = lane >> 4, m = lane & 15;
  const int ia = imin(i0 + m, L - 1);
  const int kc = imin(k0t + m, KD - 1);
  const float* rrow = rel + ((size_t)b * L + ia) * L;
  v8f acc = {};
  for (int j0 = 0; j0 < L; j0 += 4) {
    const int jj = j0 + half * 2;
    v2f a, bb;
    a.x = rrow[jj]; a.y = rrow[jj + 1];
    bb.x = q_rep[((size_t)b * L + jj) * KD + kc];
    bb.y = q_rep[((size_t)b * L + jj + 1) * KD + kc];
    acc = wmma_f32(a, bb, acc);
  }
  for (int v = 0; v < 8; ++v) {
    const int i = i0 + v + half * 8;
    const int k = k0t + m;
    if (i < L && k < KD) {
      wq[((size_t)b * L + i) * KD + k] = acc[v] * rsuminv[b * L + i];
    }
  }
}

// ---------------------------------------------------------------------------
// Kernel 5: maxpool matching with fused max/mean over j.
// One block per (b,i); 4 waves = 4 c-tiles of 16 channels.
// Factorization: mnum[j,c] = sum_k q[j,k] * (d[i,k]*M^2[c,k]).
//   B[k,n=c] = d[i,k]*M^2[c,k]  -> j-invariant, precomputed once into LDS.
//   A[m=j,k] = q[j,k]           -> staged per j-tile into LDS.
// Inner K-loop: pure LDS loads -> WMMA (no VALU multiplies).
// ---------------------------------------------------------------------------
__global__ __launch_bounds__(128) void maxpool_kernel(
    const float* __restrict__ d_rep, const float* __restrict__ q_rep,
    const float* __restrict__ mM, const float* __restrict__ rmqn,
    const float* __restrict__ rmpn, float* __restrict__ out) {
  const int bi = blockIdx.x;
  const int b = bi / L, i = bi % L;
  const int tid = threadIdx.x;
  const int lane = tid & 31, wave = tid >> 5;
  const int half = lane >> 4, n = lane & 15;
  const int c = wave * 16 + n;  // channel for B column / D column

  __shared__ float dsh[KD];        // d row
  __shared__ float bsh[64 * KP];   // d[i,k]*M^2[c,k], c = 0..63 (0 for c>=C)
  __shared__ float qsh[16 * KP];   // current j-tile of q rows

  for (int k = tid; k < KD; k += 128) dsh[k] = d_rep[(size_t)bi * KD + k];
  __syncthreads();
  for (int t = tid; t < 64 * KD; t += 128) {
    const int cidx = t / KD, k = t % KD;
    const float m = (cidx < C) ? mM[(size_t)cidx * KD + k] : 0.f;
    bsh[cidx * KP + k] = dsh[k] * m * m;
  }

  const float* bcol = bsh + (size_t)c * KP;
  const float* qrowsh = qsh + (size_t)n * KP;
  float maxAcc = -3.402823466e38f;
  float sumAcc = 0.f;

  for (int j0 = 0; j0 < L; j0 += 16) {
    __syncthreads();
    for (int t = tid; t < 16 * KD; t += 128) {
      const int r = t / KD, k = t % KD;
      const int j = imin(j0 + r, L - 1);
      qsh[r * KP + k] = q_rep[((size_t)b * L + j) * KD + k];
    }
    __syncthreads();
    v8f acc = {};
    for (int k0 = 0; k0 < KD; k0 += 4) {
      const int kk = k0 + half * 2;
      v2f a, bb;
      a.x = qrowsh[kk];
      a.y = qrowsh[kk + 1];
      bb.x = bcol[kk];
      bb.y = bcol[kk + 1];
      acc = wmma_f32(a, bb, acc);
    }
    if (c < C) {
      for (int v = 0; v < 8; ++v) {
        const int j = j0 + v + half * 8;
        if (j < L) {
          const float val = acc[v] * rmqn[((size_t)b * L + j) * C + c];
          maxAcc = fmaxf(maxAcc, val);
          sumAcc += val;
        }
      }
    }
  }
  // combine the two lane halves (rows j split across lanes 0-15 / 16-31)
  const float omax = __shfl_xor(maxAcc, 16, 32);
  const float osum = __shfl_xor(sumAcc, 16, 32);
  maxAcc = fmaxf(maxAcc, omax);
  sumAcc += osum;
  if (half == 0 && c < C) {
    const float rp = rmpn[(size_t)bi * C + c];
    float* o = out + (size_t)bi * OUTC;
    o[C + c] = maxAcc * rp;                     // maxpool max
    o[2 * C + c] = (sumAcc * (1.0f / L)) * rp;  // maxpool mean
  }
}

// ---------------------------------------------------------------------------
// Kernel 6: full matching, attentive matching, max-attentive matching.
// ---------------------------------------------------------------------------
__global__ __launch_bounds__(64) void final_kernel(
    const float* __restrict__ d_rep, const float* __restrict__ q_rep,
    const float* __restrict__ q_last, const float* __restrict__ wq,
    const int* __restrict__ argidx,
    const float* __restrict__ fM, const float* __restrict__ aM,
    const float* __restrict__ xM,
    const float* __restrict__ rfpn, const float* __restrict__ rfqn,
    const float* __restrict__ rapn, const float* __restrict__ rxpn,
    const float* __restrict__ rxqn, float* __restrict__ out) {
  const int bi = blockIdx.x;
  const int b = bi / L;
  const int tid = threadIdx.x;
  const int jstar = argidx[bi];
  __shared__ float dsh[KD], wqsh[KD], qjsh[KD], qlsh[KD];
  for (int k = tid; k < KD; k += 64) {
    dsh[k] = d_rep[(size_t)bi * KD + k];
    wqsh[k] = wq[(size_t)bi * KD + k];
    qjsh[k] = q_rep[((size_t)b * L + jstar) * KD + k];
    qlsh[k] = q_last[(size_t)b * KD + k];
  }
  __syncthreads();
  if (tid < C) {
    const int c = tid;
    const float* fr = fM + (size_t)c * KD;
    const float* ar = aM + (size_t)c * KD;
    const float* xr = xM + (size_t)c * KD;
    float accf = 0.f, acca = 0.f, accx = 0.f, wn = 0.f;
    for (int k = 0; k < KD; ++k) {
      const float xd = dsh[k];
      float m, m2;
      m = fr[k]; m2 = m * m; accf += xd * qlsh[k] * m2;
      m = ar[k]; m2 = m * m;
      acca += xd * wqsh[k] * m2;
      wn += wqsh[k] * wqsh[k] * m2;
      m = xr[k]; m2 = m * m; accx += xd * qjsh[k] * m2;
    }
    float* o = out + (size_t)bi * OUTC;
    const size_t oc = (size_t)bi * C + c;
    o[c] = accf * rfpn[oc] * rfqn[(size_t)b * C + c];
    o[3 * C + c] = acca * rapn[oc] * rnorm(wn);
    o[4 * C + c] = accx * rxpn[oc] * rxqn[((size_t)b * L + jstar) * C + c];
  }
}

// ---------------------------------------------------------------------------
// Workspace layout (floats)
// ---------------------------------------------------------------------------
#define OFF_REL    ((size_t)0)
#define OFF_QN     (OFF_REL + (size_t)NB * L * L)
#define OFF_DN     (OFF_QN + (size_t)NB * L)
#define OFF_RSUM   (OFF_DN + (size_t)NB * L)
#define OFF_MQN    (OFF_RSUM + (size_t)NB * L)
#define OFF_MPN    (OFF_MQN + (size_t)NB * L * C)
#define OFF_FPN    (OFF_MPN + (size_t)NB * L * C)
#define OFF_FQN    (OFF_FPN + (size_t)NB * L * C)
#define OFF_APN    (OFF_FQN + (size_t)NB * C)
#define OFF_XPN    (OFF_APN + (size_t)NB * L * C)
#define OFF_XQN    (OFF_XPN + (size_t)NB * L * C)
#define OFF_WQ     (OFF_XQN + (size_t)NB * L * C)
#define OFF_IDX    (OFF_WQ + (size_t)NB * L * KD)

extern "C" void kernel_launch(void* const* d_in, const int* in_sizes, int n_in,
                              void* d_out, int out_size, void* d_ws,
                              size_t ws_size, hipStream_t stream) {
  const float* q_rep = (const float*)d_in[0];
  const float* q_last = (const float*)d_in[1];
  const float* q_mask = (const float*)d_in[2];
  const float* d_rep = (const float*)d_in[3];
  // d_in[4] = d_last (unused by reference)
  const float* d_mask = (const float*)d_in[5];
  const float* full_M = (const float*)d_in[6];
  const float* maxpool_M = (const float*)d_in[7];
  const float* att_M = (const float*)d_in[8];
  const float* max_att_M = (const float*)d_in[9];
  float* out = (float*)d_out;
  float* ws = (float*)d_ws;

  float* rel = ws + OFF_REL;
  float* rqn = ws + OFF_QN;
  float* rdn = ws + OFF_DN;
  float* rsuminv = ws + OFF_RSUM;
  float* rmqn = ws + OFF_MQN;
  float* rmpn = ws + OFF_MPN;
  float* rfpn = ws + OFF_FPN;
  float* rfqn = ws + OFF_FQN;
  float* rapn = ws + OFF_APN;
  float* rxpn = ws + OFF_XPN;
  float* rxqn = ws + OFF_XQN;
  float* wq = ws + OFF_WQ;
  int* idx = (int*)(ws + OFF_IDX);

  d_norms_kernel<<<NB * L, 64, 0, stream>>>(d_rep, full_M, maxpool_M, att_M,
                                            max_att_M, rdn, rfpn, rmpn, rapn,
                                            rxpn);
  q_norms_kernel<<<NB * L, 64, 0, stream>>>(q_rep, maxpool_M, max_att_M, rqn,
                                            rmqn, rxqn);
  qlast_norms_kernel<<<NB, 64, 0, stream>>>(q_last, full_M, rfqn);
  rel_kernel<<<dim3(NT * NT, NB), 32, 0, stream>>>(d_rep, q_rep, q_mask,
                                                   d_mask, rqn, rdn, rel);
  rel_reduce_kernel<<<NB * L, 64, 0, stream>>>(rel, rsuminv, idx);
  wq_kernel<<<dim3(NT * NT, NB), 32, 0, stream>>>(rel, q_rep, rsuminv, wq);
  maxpool_kernel<<<NB * L, 128, 0, stream>>>(d_rep, q_rep, maxpool_M, rmqn,
                                             rmpn, out);
  final_kernel<<<NB * L, 64, 0, stream>>>(d_rep, q_rep, q_last, wq, idx,
                                          full_M, att_M, max_att_M, rfpn, rfqn,
                                          rapn, rxpn, rxqn, out);
}